// RankerCosEmbLoss_28011776704955
// MI455X (gfx1250) — compile-verified
//
#include <hip/hip_runtime.h>

// CDNA5 / gfx1250: wave32, WMMA-based cross-lane reductions.

typedef __attribute__((ext_vector_type(2))) float v2f;
typedef __attribute__((ext_vector_type(4))) float v4f;
typedef __attribute__((ext_vector_type(4))) int   v4i;
typedef __attribute__((ext_vector_type(8))) float v8f;

#define N_DOCS  4096
#define N_CANDS 8192

// Full-wave (32-lane) sum using V_WMMA_F32_16X16X4_F32 with a ones B-matrix.
// A-matrix 16x4 f32 layout: VGPR0 = {lanes 0-15: K=0, lanes 16-31: K=2},
// VGPR1 = {K=1, K=3} (set to zero). With B==1: D[m][n] = p_m + p_{m+16}.
// Lane L<16 holds D rows 0..7 across the 8 D VGPRs, lanes >=16 hold rows 8..15,
// so (sum of 8 D VGPRs) + shfl_xor(16) = full 32-lane sum, in every lane.
__device__ __forceinline__ float wave_sum_wmma(float v) {
    v2f a; a[0] = v;    a[1] = 0.0f;
    v2f b; b[0] = 1.0f; b[1] = 1.0f;
    v8f c = {};
    v8f d = __builtin_amdgcn_wmma_f32_16x16x4_f32(
        /*neg_a=*/false, a, /*neg_b=*/false, b,
        /*c_mod=*/(short)0, c, /*reuse_a=*/false, /*reuse_b=*/false);
    float s = d[0] + d[1] + d[2] + d[3] + d[4] + d[5] + d[6] + d[7];
    s += __shfl_xor(s, 16, 32);
    return s;
}

// One 256-thread block per row. Streams 64 KB/row (f32 + i32) with
// non-temporal 128-bit loads; writes per-row loss terms to workspace.
__global__ __launch_bounds__(256)
void RankerCosEmbLoss_row_kernel(const v4f* __restrict__ cos4,
                                 const v4i* __restrict__ msk4,
                                 float* __restrict__ ws) {
    const int row = blockIdx.x;
    const int tid = threadIdx.x;
    const long base = (long)row * (N_CANDS / 4);

    float st = 0.0f;  // sum of (1 - cos) over mask==1
    float sn = 0.0f;  // sum of max(cos, 0) over mask==0
    float ct = 0.0f;  // count of mask==1

#pragma unroll
    for (int i = 0; i < 8; ++i) {
        const long idx = base + (long)(i * 256 + tid);
        v4f c4 = __builtin_nontemporal_load(&cos4[idx]);   // TH=NT: single-use stream
        v4i m4 = __builtin_nontemporal_load(&msk4[idx]);
#pragma unroll
        for (int j = 0; j < 4; ++j) {
            const float cv = c4[j];
            const bool  m  = (m4[j] != 0);
            st += m ? (1.0f - cv) : 0.0f;
            sn += m ? 0.0f : fmaxf(cv, 0.0f);
            ct += m ? 1.0f : 0.0f;
        }
    }

    // Wave-level reductions via WMMA (uniform control flow, EXEC all-1s).
    const float stw = wave_sum_wmma(st);
    const float snw = wave_sum_wmma(sn);
    const float ctw = wave_sum_wmma(ct);

    __shared__ float sm[3][8];
    const int wave = tid >> 5;
    const int lane = tid & 31;
    if (lane == 0) { sm[0][wave] = stw; sm[1][wave] = snw; sm[2][wave] = ctw; }
    __syncthreads();

    if (tid == 0) {
        float S = 0.0f, Nn = 0.0f, C = 0.0f;
#pragma unroll
        for (int w = 0; w < 8; ++w) { S += sm[0][w]; Nn += sm[1][w]; C += sm[2][w]; }
        const float Cn = (float)N_CANDS - C;
        ws[row]          = S / C;     // per-row target term
        ws[N_DOCS + row] = Nn / Cn;   // per-row non-target term
    }
}

// Single block: mean over 4096 rows for both terms, emit (loss, loss_tgt, loss_non).
__global__ __launch_bounds__(256)
void RankerCosEmbLoss_final_kernel(const float* __restrict__ ws,
                                   float* __restrict__ out) {
    const int tid = threadIdx.x;
    float pt = 0.0f, pn = 0.0f;
#pragma unroll
    for (int i = 0; i < N_DOCS / 256; ++i) {
        pt += ws[i * 256 + tid];
        pn += ws[N_DOCS + i * 256 + tid];
    }
    const float tw = wave_sum_wmma(pt);
    const float nw = wave_sum_wmma(pn);

    __shared__ float sm[2][8];
    const int wave = tid >> 5;
    const int lane = tid & 31;
    if (lane == 0) { sm[0][wave] = tw; sm[1][wave] = nw; }
    __syncthreads();

    if (tid == 0) {
        float T = 0.0f, Nt = 0.0f;
#pragma unroll
        for (int w = 0; w < 8; ++w) { T += sm[0][w]; Nt += sm[1][w]; }
        const float loss_tgt = T  / (float)N_DOCS;
        const float loss_non = Nt / (float)N_DOCS;
        out[0] = 0.5f * (loss_tgt + loss_non);
        out[1] = loss_tgt;
        out[2] = loss_non;
    }
}

extern "C" void kernel_launch(void* const* d_in, const int* in_sizes, int n_in,
                              void* d_out, int out_size, void* d_ws, size_t ws_size,
                              hipStream_t stream) {
    const v4f* cos4 = (const v4f*)d_in[0];   // cos_pred, float32 [4096, 8192]
    const v4i* msk4 = (const v4i*)d_in[1];   // mask_gt,   int32   [4096, 8192]
    float* ws  = (float*)d_ws;               // 2 * 4096 floats of per-row terms
    float* out = (float*)d_out;              // 3 floats: loss, loss_tgt, loss_non

    RankerCosEmbLoss_row_kernel<<<N_DOCS, 256, 0, stream>>>(cos4, msk4, ws);
    RankerCosEmbLoss_final_kernel<<<1, 256, 0, stream>>>(ws, out);
}